// LightGCN_73882027425875
// MI455X (gfx1250) — compile-verified
//
#include <hip/hip_runtime.h>
#include <cstdint>

// ---------------------------------------------------------------------------
// LightGCN: 3x SpMM over 3.2M random edges, 200K x 64 f32 embeddings.
// Memory/atomic-bound (AI ~0.25 flop/byte): WMMA not applicable.
// CDNA5 paths: global_load_async_to_lds_b32 (th:TH_LOAD_NT) + s_wait_asynccnt
// for edge staging; native f32 global atomics (global_atomic_add_f32).
// Both embedding tables (2 x 51.2MB) live in the 192MB L2; the edge stream is
// marked non-temporal so it does not evict them.
// ---------------------------------------------------------------------------

namespace lgcn {
constexpr int NUM_USERS = 100000;
constexpr int NUM_ITEMS = 100000;
constexpr int N_NODES   = NUM_USERS + NUM_ITEMS;  // 200000
constexpr int EMB       = 64;
constexpr int N_EDGES   = 3200000;
constexpr int TILE      = 256;                    // edges per block
constexpr int NBLK_EDGE = N_EDGES / TILE;         // 12500, exact
constexpr size_t NODE_FLOATS = (size_t)N_NODES * EMB;     // 12.8M
constexpr size_t NODE_QUADS  = NODE_FLOATS / 4;           // 3.2M
constexpr int NBLK_ELT  = (int)(NODE_QUADS / 256);        // 12500, exact
constexpr float INV3 = 1.0f / 3.0f;
}

// ego0 = concat(user, item); ego1 = 0; acc(out) = 0
__global__ void lgcn_init(const float4* __restrict__ user,
                          const float4* __restrict__ item,
                          float4* __restrict__ ego0,
                          float4* __restrict__ ego1,
                          float4* __restrict__ acc) {
    const size_t i  = (size_t)blockIdx.x * blockDim.x + threadIdx.x;  // < 3.2M
    constexpr size_t UQ = (size_t)lgcn::NUM_USERS * lgcn::EMB / 4;    // 1.6M
    const float4 v = (i < UQ) ? user[i] : item[i - UQ];
    const float4 z = make_float4(0.f, 0.f, 0.f, 0.f);
    ego0[i] = v;
    ego1[i] = z;
    acc[i]  = z;
}

// acc += src / 3 ; optionally zero zbuf (prepares next layer's output buffer)
__global__ void lgcn_accum(float4* __restrict__ acc,
                           const float4* __restrict__ src,
                           float4* __restrict__ zbuf,
                           int doZero) {
    const size_t i = (size_t)blockIdx.x * blockDim.x + threadIdx.x;
    float4 a = acc[i];
    const float4 s = src[i];
    a.x = fmaf(s.x, lgcn::INV3, a.x);
    a.y = fmaf(s.y, lgcn::INV3, a.y);
    a.z = fmaf(s.z, lgcn::INV3, a.z);
    a.w = fmaf(s.w, lgcn::INV3, a.w);
    acc[i] = a;
    if (doZero) zbuf[i] = make_float4(0.f, 0.f, 0.f, 0.f);
}

// y[row] += w * x[col].  One block = 256 edges; one wave = 32 edges, one edge
// at a time with the full wave covering all 64 floats (float2 per lane).
__global__ void lgcn_spmm(const float* __restrict__ x,
                          float* __restrict__ y,
                          const int* __restrict__ row,
                          const int* __restrict__ col,
                          const float* __restrict__ w) {
    __shared__ int   s_row[lgcn::TILE];
    __shared__ int   s_col[lgcn::TILE];
    __shared__ float s_w[lgcn::TILE];

    const int t = threadIdx.x;
    const int e = blockIdx.x * lgcn::TILE + t;   // exact, no tail

    // --- Stage this block's edge tile into LDS via async global->LDS copies.
    // Generic shared-space pointers carry the LDS byte offset in their low 32
    // bits (ISA 10.2), which is what the vdst VGPR of
    // global_load_async_to_lds_b32 expects.  The edge stream is touch-once:
    // mark it non-temporal so it does not evict the embedding tables from L2.
    {
        const unsigned offR = (unsigned)(uintptr_t)&s_row[t];
        const unsigned offC = (unsigned)(uintptr_t)&s_col[t];
        const unsigned offW = (unsigned)(uintptr_t)&s_w[t];
        const int*   pr = row + e;
        const int*   pc = col + e;
        const float* pw = w + e;
        asm volatile("global_load_async_to_lds_b32 %0, %1, off th:TH_LOAD_NT"
                     :: "v"(offR), "v"(pr) : "memory");
        asm volatile("global_load_async_to_lds_b32 %0, %1, off th:TH_LOAD_NT"
                     :: "v"(offC), "v"(pc) : "memory");
        asm volatile("global_load_async_to_lds_b32 %0, %1, off th:TH_LOAD_NT"
                     :: "v"(offW), "v"(pw) : "memory");
        // Each wave reads back only entries staged by its own lanes, so a
        // per-wave asynccnt wait is sufficient (no workgroup barrier).
        asm volatile("s_wait_asynccnt 0" ::: "memory");
    }

    const int lane  = t & 31;
    const int wbase = (t >> 5) << 5;   // this wave's 32 edges in the tile

#pragma unroll 4
    for (int i = 0; i < 32; ++i) {
        const int   idx = wbase + i;
        const int   r   = s_row[idx];          // LDS broadcast reads
        const int   c   = s_col[idx];
        const float wt  = s_w[idx];

        // Coalesced 256B gather: wave covers 64 floats, float2 per lane.
        const float2 xv = *reinterpret_cast<const float2*>(
            x + (size_t)c * lgcn::EMB + lane * 2);

        float* yp = y + (size_t)r * lgcn::EMB + lane * 2;
        // Native no-return f32 atomics into L2 (dest table fits in 192MB L2).
        unsafeAtomicAdd(yp,     xv.x * wt);
        unsafeAtomicAdd(yp + 1, xv.y * wt);
    }
}

extern "C" void kernel_launch(void* const* d_in, const int* in_sizes, int n_in,
                              void* d_out, int out_size, void* d_ws, size_t ws_size,
                              hipStream_t stream) {
    using namespace lgcn;
    const float* user_emb = (const float*)d_in[0];
    const float* item_emb = (const float*)d_in[1];
    const float* edge_w   = (const float*)d_in[2];
    const int*   edge_row = (const int*)d_in[3];
    const int*   edge_col = (const int*)d_in[4];
    float* out = (float*)d_out;

    float* ego0 = (float*)d_ws;
    float* ego1 = ego0 + NODE_FLOATS;

    lgcn_init<<<NBLK_ELT, 256, 0, stream>>>(
        (const float4*)user_emb, (const float4*)item_emb,
        (float4*)ego0, (float4*)ego1, (float4*)out);

    float* src = ego0;   // layer input (read)
    float* dst = ego1;   // layer output (pre-zeroed)
    for (int layer = 0; layer < 3; ++layer) {
        lgcn_spmm<<<NBLK_EDGE, TILE, 0, stream>>>(src, dst, edge_row, edge_col, edge_w);
        // out += dst/3 ; re-zero `src` so it can serve as the next layer's
        // output buffer (skipped on the last layer: lgcn_init re-establishes
        // all invariants at the start of every call, keeping replays
        // deterministic).
        lgcn_accum<<<NBLK_ELT, 256, 0, stream>>>((float4*)out, (const float4*)dst,
                                                 (float4*)src, layer < 2 ? 1 : 0);
        float* tmp = src; src = dst; dst = tmp;
    }
}